// LogSignature_13511967113633
// MI455X (gfx1250) — compile-verified
//
#include <hip/hip_runtime.h>

typedef __attribute__((ext_vector_type(2))) float v2f;
typedef __attribute__((ext_vector_type(8))) float v8f;

#define CD 8
#define TT 512
#define NSEG 511          // T-1 increments
#define NTHREADS 128      // 4 wave32 per block, one block per batch

// Log-signature depth-4, C=8. One workgroup per batch element.
// Level-4 Chen update done as V_WMMA_F32_16X16X4_F32 rank-4 updates with
// persistent D accumulators in VGPRs (32 tiles of 16x16, 8 per wave).
__global__ __launch_bounds__(NTHREADS)
void logsig_kernel(const float* __restrict__ path, float* __restrict__ out) {
    __shared__ float dxs[NSEG * CD];   // all segment increments (16352 B)
    __shared__ float S3s[512];         // level-3 state
    __shared__ float S2b[2][64];       // level-2 state (ping-pong)
    __shared__ float S1b[2][8];        // level-1 state (ping-pong)
    __shared__ float S4s[4096];        // level-4 dump for epilogue

    const int b    = blockIdx.x;
    const int tid  = threadIdx.x;
    const int lane = tid & 31;
    const int w    = tid >> 5;     // wave id 0..3
    const int Ln   = lane & 15;    // row within 16x16 tile
    const int half = lane >> 4;    // 0: A cols {0,1}; 1: A cols {2,3}

    const float* pb = path + (size_t)b * (TT * CD);

    // Precompute increments dx[t][c] = path[t+1][c] - path[t][c]
    for (int e = tid; e < NSEG * CD; e += NTHREADS)
        dxs[e] = pb[e + 8] - pb[e];
    for (int e = tid; e < 512; e += NTHREADS) S3s[e] = 0.0f;
    if (tid < 64) S2b[0][tid] = 0.0f;
    if (tid < 8)  S1b[0][tid] = 0.0f;

    // Persistent level-4 accumulators: wave w owns tiles t = 8w..8w+7,
    // tile t covers S4 rows 16t..16t+15 (of the 512x8 reshape).
    v8f acc[8];
    const v8f vzero = {0.f, 0.f, 0.f, 0.f, 0.f, 0.f, 0.f, 0.f};
#pragma unroll
    for (int q = 0; q < 8; ++q) acc[q] = vzero;

    int cur = 0;
    for (int t = 0; t < NSEG; ++t) {
        __syncthreads();   // prior step's S3/S2/S1 writes (and dx init) visible
        const float* d = &dxs[t * CD];
        const float dk = d[Ln & 7];   // k index = r&7 = (16*tile+Ln)&7 = Ln&7

        // B fragment: all 4 K-rows equal d (cols 8..15 zero) -> K-row
        // permutation irrelevant. Coefficients folded into A columns.
        const float bv = (Ln < 8) ? d[Ln] : 0.0f;
        v2f bf; bf[0] = bv; bf[1] = bv;

        float s3n[8];
#pragma unroll
        for (int q = 0; q < 8; ++q) {
            const int tt = w * 8 + q;
            const int r  = tt * 16 + Ln;          // row of S4 (512x8 reshape)
            const int i  = r >> 6;
            const int j  = (r >> 3) & 7;
            const float s2v = S2b[cur][(i << 3) | j];
            const float s1v = S1b[cur][i];
            const float di = d[i], dj = d[j];
            const float p1 = s2v * dk;            // (S2 (x) d)(r)
            const float p2 = s1v * dj * dk;       // (S1 (x) d (x) d)(r)
            const float p3 = di * dj * dk;        // (d (x) d (x) d)(r)
            const float s3v = S3s[r];

            // A(16x4): cols = [S3, p1/2, p2/6, p3/24]; lanes 0-15 hold cols
            // {0,1}, lanes 16-31 hold cols {2,3} (VGPR0/VGPR1 per ISA layout).
            v2f af;
            af[0] = half ? (1.0f / 6.0f)  * p2 : s3v;
            af[1] = half ? (1.0f / 24.0f) * p3 : 0.5f * p1;

            acc[q] = __builtin_amdgcn_wmma_f32_16x16x4_f32(
                false, af, false, bf, (short)0, acc[q], false, false);

            // Level-3 Chen update (uses OLD S2/S1 via p's)
            s3n[q] = s3v + p1 + 0.5f * p2 + (1.0f / 6.0f) * p3;
        }
        // In-place S3 update: each row written only by the lane that read it.
        if (half == 0) {
#pragma unroll
            for (int q = 0; q < 8; ++q)
                S3s[(w * 8 + q) * 16 + Ln] = s3n[q];
        }
        // Level-2 / level-1 Chen updates into ping-pong buffers.
        const int nxt = cur ^ 1;
        if (tid < 64) {
            const int i2 = tid >> 3, j2 = tid & 7;
            S2b[nxt][tid] = S2b[cur][tid] + S1b[cur][i2] * d[j2]
                          + 0.5f * d[i2] * d[j2];
        } else if (tid < 72) {
            const int c = tid - 64;
            S1b[nxt][c] = S1b[cur][c] + d[c];
        }
        cur = nxt;
    }

    __syncthreads();
    // Dump WMMA accumulators to LDS. D layout: D(m,n) in VGPR (m&7),
    // lane = (m>=8 ? 16 : 0) + n. Only n<8 columns are real data.
    if (Ln < 8) {
#pragma unroll
        for (int q = 0; q < 8; ++q) {
            const int tt = w * 8 + q;
#pragma unroll
            for (int v = 0; v < 8; ++v) {
                const int m = v + (half << 3);
                S4s[(tt * 16 + m) * 8 + Ln] = acc[q][v];
            }
        }
    }
    __syncthreads();

    // log(1+S) truncated at depth 4, written as concat(L1,L2,L3,L4).
    float* ob = out + (size_t)b * 4680;
    const float* S1f = S1b[cur];
    const float* S2f = S2b[cur];

    if (tid < 8) ob[tid] = S1f[tid];
    if (tid < 64) {
        const int i = tid >> 3, j = tid & 7;
        ob[8 + tid] = S2f[tid] - 0.5f * S1f[i] * S1f[j];
    }
    for (int e = tid; e < 512; e += NTHREADS) {
        const int i = e >> 6, j = (e >> 3) & 7, k = e & 7;
        const float v3 = S3s[e]
            - 0.5f * (S1f[i] * S2f[(j << 3) | k] + S2f[(i << 3) | j] * S1f[k])
            + (1.0f / 3.0f) * S1f[i] * S1f[j] * S1f[k];
        ob[72 + e] = v3;
    }
    for (int e = tid; e < 4096; e += NTHREADS) {
        const int i = e >> 9, j = (e >> 6) & 7, k = (e >> 3) & 7, l = e & 7;
        const float s1i = S1f[i], s1j = S1f[j], s1k = S1f[k], s1l = S1f[l];
        const float v4 = S4s[e]
            - 0.5f * ( s1i * S3s[e & 511]
                     + S2f[(i << 3) | j] * S2f[(k << 3) | l]
                     + S3s[e >> 3] * s1l )
            + (1.0f / 3.0f) * ( s1i * s1j * S2f[(k << 3) | l]
                              + s1i * S2f[(j << 3) | k] * s1l
                              + S2f[(i << 3) | j] * s1k * s1l )
            - 0.25f * s1i * s1j * s1k * s1l;
        ob[584 + e] = v4;
    }
}

extern "C" void kernel_launch(void* const* d_in, const int* in_sizes, int n_in,
                              void* d_out, int out_size, void* d_ws, size_t ws_size,
                              hipStream_t stream) {
    const float* path = (const float*)d_in[0];
    // d_in[1] is depth (==4), baked into the kernel.
    float* out = (float*)d_out;
    const int B = in_sizes[0] / (TT * CD);   // 128
    logsig_kernel<<<dim3(B), dim3(NTHREADS), 0, stream>>>(path, out);
}